// Model_32925219291401
// MI455X (gfx1250) — compile-verified
//
#include <hip/hip_runtime.h>
#include <hip/hip_bf16.h>

typedef __attribute__((ext_vector_type(2))) float v2f;
typedef __attribute__((ext_vector_type(4))) float v4f;
typedef __attribute__((ext_vector_type(8))) float v8f;

// ---------------------------------------------------------------------------
// Degree / normalization
// ---------------------------------------------------------------------------
__global__ void k_init_deg(int* __restrict__ deg, int n) {
  int i = blockIdx.x * blockDim.x + threadIdx.x;
  if (i < n) deg[i] = 1;  // self-loop contributes 1
}

__global__ void k_count_deg(const int* __restrict__ dst, int* __restrict__ deg, int E) {
  int e = blockIdx.x * blockDim.x + threadIdx.x;
  if (e < E) atomicAdd(&deg[dst[e]], 1);
}

__global__ void k_dinv(const int* __restrict__ deg, float* __restrict__ dinv, int n) {
  int i = blockIdx.x * blockDim.x + threadIdx.x;
  if (i < n) dinv[i] = 1.0f / sqrtf((float)deg[i]);  // deg >= 1 always
}

// ---------------------------------------------------------------------------
// Dense GEMM via V_WMMA_F32_16X16X4_F32, one wave per 16x16 C tile.
// A: [M,K] row-major, B: [K,N] row-major, C: [M,N] row-major.
// Requires M%16==0, N%16==0, K%4==0 (true for all calls here), so every wave
// runs with EXEC all-ones as WMMA requires.
// ---------------------------------------------------------------------------
__global__ void k_gemm_wmma_f32(const float* __restrict__ A,
                                const float* __restrict__ B,
                                float* __restrict__ C,
                                int M, int N, int K) {
  int wave = (int)((blockIdx.x * blockDim.x + threadIdx.x) >> 5);
  int lane = threadIdx.x & 31;
  int tilesN = N >> 4;
  int total = (M >> 4) * tilesN;
  if (wave >= total) return;  // wave-uniform

  int tm = wave / tilesN;
  int tn = wave - tm * tilesN;
  int row0 = tm << 4;
  int col0 = tn << 4;
  int l16 = lane & 15;
  int hi  = lane >> 4;  // 0 or 1

  const float* __restrict__ arow = A + (size_t)(row0 + l16) * K;
  v8f acc = {};
  for (int k = 0; k < K; k += 4) {
    int ka = k + 2 * hi;                 // K index for VGPR slot 0 of this lane
    v2f a, b;
    a.x = arow[ka];
    a.y = arow[ka + 1];
    b.x = B[(size_t)ka * N + col0 + l16];
    b.y = B[(size_t)(ka + 1) * N + col0 + l16];
    acc = __builtin_amdgcn_wmma_f32_16x16x4_f32(
        /*neg_a=*/false, a, /*neg_b=*/false, b,
        /*c_mod=*/(short)0, acc, /*reuse_a=*/false, /*reuse_b=*/false);
  }
#pragma unroll
  for (int r = 0; r < 8; ++r) {
    int row = row0 + r + 8 * hi;
    C[(size_t)row * N + col0 + l16] = acc[r];
  }
}

// ---------------------------------------------------------------------------
// agg[i,f4] = bias[f4] + dinv[i]^2 * h[i,f4]   (bias + self-loop), float4-wide
// n4 = Nn * F/4 total float4 elements, F4 = F/4
// ---------------------------------------------------------------------------
__global__ void k_init_agg4(const v4f* __restrict__ h,
                            const float* __restrict__ dinv,
                            const v4f* __restrict__ bias,
                            v4f* __restrict__ agg, int n4, int F4) {
  int i = blockIdx.x * blockDim.x + threadIdx.x;
  if (i >= n4) return;
  int node = i / F4;
  int f4 = i - node * F4;
  float di = dinv[node];
  float s = di * di;
  v4f hv = h[i];
  v4f bv = bias[f4];
  v4f r;
  r.x = bv.x + s * hv.x;
  r.y = bv.y + s * hv.y;
  r.z = bv.z + s * hv.z;
  r.w = bv.w + s * hv.w;
  agg[i] = r;
}

// ---------------------------------------------------------------------------
// Edge scatter: one wave per edge; lanes stride the feature dim in float4s.
// agg[dst,f] += dinv[src]*dinv[dst] * h[src,f]
// Gather is a b128 load per lane (512B coalesced per wave-iteration);
// atomics are scalar global_atomic_add_f32 (no packed f32 atomic exists).
// ---------------------------------------------------------------------------
__global__ void k_scatter(const int* __restrict__ srcs,
                          const int* __restrict__ dsts,
                          const float* __restrict__ dinv,
                          const float* __restrict__ h,
                          float* __restrict__ agg, int E, int F) {
  long long gid = (long long)blockIdx.x * blockDim.x + threadIdx.x;
  int e = (int)(gid >> 5);
  int lane = (int)(gid & 31);
  if (e >= E) return;
  int s = srcs[e];
  int d = dsts[e];
  float nrm = dinv[s] * dinv[d];
  const v4f* __restrict__ hs4 = (const v4f*)(h + (size_t)s * F);
  float* ad = agg + (size_t)d * F;
  int F4 = F >> 2;
  for (int f4 = lane; f4 < F4; f4 += 32) {
    v4f v = hs4[f4];
    int f = f4 << 2;
    unsafeAtomicAdd(&ad[f + 0], nrm * v.x);
    unsafeAtomicAdd(&ad[f + 1], nrm * v.y);
    unsafeAtomicAdd(&ad[f + 2], nrm * v.z);
    unsafeAtomicAdd(&ad[f + 3], nrm * v.w);
  }
}

__global__ void k_relu4(v4f* __restrict__ x, int n4) {
  int i = blockIdx.x * blockDim.x + threadIdx.x;
  if (i >= n4) return;
  v4f v = x[i];
  v.x = fmaxf(v.x, 0.0f);
  v.y = fmaxf(v.y, 0.0f);
  v.z = fmaxf(v.z, 0.0f);
  v.w = fmaxf(v.w, 0.0f);
  x[i] = v;
}

// ---------------------------------------------------------------------------
// Launcher
// ---------------------------------------------------------------------------
extern "C" void kernel_launch(void* const* d_in, const int* in_sizes, int n_in,
                              void* d_out, int out_size, void* d_ws, size_t ws_size,
                              hipStream_t stream) {
  constexpr int IN_CH = 128, HID = 256, OUT_CH = 128;

  const float* x  = (const float*)d_in[0];
  const int*   ei = (const int*)d_in[1];
  const float* W1 = (const float*)d_in[2];
  const float* b1 = (const float*)d_in[3];
  const float* W2 = (const float*)d_in[4];
  const float* b2 = (const float*)d_in[5];
  float* out = (float*)d_out;

  const int Nn = in_sizes[0] / IN_CH;   // 10000
  const int E  = in_sizes[1] / 2;       // 640000

  // Workspace carve-up (all offsets 16B aligned for Nn=10000)
  char* ws = (char*)d_ws;
  int*   deg  = (int*)ws;                                   // Nn ints
  float* dinv = (float*)(ws + (size_t)Nn * sizeof(int));    // Nn floats
  float* h    = (float*)(ws + 2 * (size_t)Nn * sizeof(int));// Nn*HID floats
  float* agg  = h + (size_t)Nn * HID;                       // Nn*HID floats

  const int T = 256;

  // 1) degrees + normalization
  k_init_deg<<<(Nn + T - 1) / T, T, 0, stream>>>(deg, Nn);
  k_count_deg<<<(E + T - 1) / T, T, 0, stream>>>(ei + E, deg, E);
  k_dinv<<<(Nn + T - 1) / T, T, 0, stream>>>(deg, dinv, Nn);

  // 2) layer 1 GEMM: h = x @ W1   [Nn,128]x[128,256]
  {
    int waves = (Nn / 16) * (HID / 16);
    int threads = waves * 32;
    k_gemm_wmma_f32<<<(threads + T - 1) / T, T, 0, stream>>>(x, W1, h, Nn, HID, IN_CH);
  }

  // 3) aggregate layer 1
  {
    int n4 = Nn * (HID / 4);
    k_init_agg4<<<(n4 + T - 1) / T, T, 0, stream>>>((const v4f*)h, dinv, (const v4f*)b1,
                                                    (v4f*)agg, n4, HID / 4);
  }
  {
    long long threads = (long long)E * 32;
    k_scatter<<<(unsigned)((threads + T - 1) / T), T, 0, stream>>>(ei, ei + E, dinv, h, agg, E, HID);
  }
  k_relu4<<<((Nn * HID / 4) + T - 1) / T, T, 0, stream>>>((v4f*)agg, Nn * HID / 4);

  // 4) layer 2 GEMM: h2 = relu(agg) @ W2  [Nn,256]x[256,128]  (reuse h buffer)
  {
    int waves = (Nn / 16) * (OUT_CH / 16);
    int threads = waves * 32;
    k_gemm_wmma_f32<<<(threads + T - 1) / T, T, 0, stream>>>(agg, W2, h, Nn, OUT_CH, HID);
  }

  // 5) aggregate layer 2 directly into d_out
  {
    int n4 = Nn * (OUT_CH / 4);
    k_init_agg4<<<(n4 + T - 1) / T, T, 0, stream>>>((const v4f*)h, dinv, (const v4f*)b2,
                                                    (v4f*)out, n4, OUT_CH / 4);
  }
  {
    long long threads = (long long)E * 32;
    k_scatter<<<(unsigned)((threads + T - 1) / T), T, 0, stream>>>(ei, ei + E, dinv, h, out, E, OUT_CH);
  }
  k_relu4<<<((Nn * OUT_CH / 4) + T - 1) / T, T, 0, stream>>>((v4f*)out, Nn * OUT_CH / 4);
}